// TGN_40166534152234
// MI455X (gfx1250) — compile-verified
//
#include <hip/hip_runtime.h>
#include <hip/hip_bf16.h>

// Problem constants (match reference)
#define NNODES 25000
#define NEDGES 400000
#define FDIM   128      // IN = TDIM = OUT = 128
#define TROWS  50000
#define DELTAT 0.2f     // MAXT / ROWS = 10000 / 50000

typedef __attribute__((ext_vector_type(2))) float v2f;
typedef __attribute__((ext_vector_type(8))) float v8f;

// ---------------------------------------------------------------------------
// order-preserving float <-> uint encoding for atomicMax on floats
// ---------------------------------------------------------------------------
__device__ __forceinline__ unsigned fenc(float f) {
  unsigned u = __float_as_uint(f);
  return (u >> 31) ? ~u : (u | 0x80000000u);
}
__device__ __forceinline__ float fdec(unsigned u) {
  u = (u >> 31) ? (u & 0x7fffffffu) : ~u;
  return __uint_as_float(u);
}

// ---------------------------------------------------------------------------
// init: zero output accumulator, softmax sums, and max-encodings
// ---------------------------------------------------------------------------
__global__ __launch_bounds__(256) void init_kernel(float* __restrict__ out,
                                                   float* __restrict__ s,
                                                   unsigned* __restrict__ menc) {
  int i = blockIdx.x * 256 + threadIdx.x;
  if (i < NNODES * FDIM) out[i] = 0.0f;
  if (i < NNODES) { s[i] = 0.0f; menc[i] = 0u; }
}

// ---------------------------------------------------------------------------
// Fused GEMM: C_j[M,128] = A[M,128] @ B_j[128,128]  for j < NB  (fp32 WMMA)
// One wave owns a 16x16 output tile; 8 waves/block span N = 128.
// A-layout  (16x4 f32): lane<16 -> M=lane, regs hold K={k0,k0+1};
//                       lane>=16 -> M=lane-16, regs hold K={k0+2,k0+3}.
// B-layout  (4x16 f32): mirrored (N = lane&15, K pair picked by lane>>4).
// D-layout  (16x16 f32): acc[r] = D[m0 + r + 8*(lane>>4)][n0 + (lane&15)].
// ---------------------------------------------------------------------------
template <int NB>
__global__ __launch_bounds__(256) void gemm_wmma_k128(
    const float* __restrict__ A, int M,
    const float* __restrict__ B0, const float* __restrict__ B1,
    const float* __restrict__ B2,
    float* __restrict__ C0, float* __restrict__ C1, float* __restrict__ C2) {
  const int lane = threadIdx.x & 31;
  const int wave = threadIdx.x >> 5;
  const int row  = lane & 15;
  const int half = lane >> 4;
  const int m0   = blockIdx.x * 16;
  const int n0   = wave * 16;

  const int ma = min(m0 + row, M - 1);           // clamp reads; stores guarded
  const float* __restrict__ Ar = A + (size_t)ma * FDIM;

  v8f acc0 = {}; v8f acc1 = {}; v8f acc2 = {};

#pragma unroll 8
  for (int k0 = 0; k0 < FDIM; k0 += 4) {
    const int ka = k0 + half * 2;
    v2f a; a.x = Ar[ka]; a.y = Ar[ka + 1];
    {
      v2f b; b.x = B0[(size_t)ka * FDIM + n0 + row];
             b.y = B0[(size_t)(ka + 1) * FDIM + n0 + row];
      acc0 = __builtin_amdgcn_wmma_f32_16x16x4_f32(false, a, false, b,
                                                   (short)0, acc0, false, false);
    }
    if (NB > 1) {
      v2f b; b.x = B1[(size_t)ka * FDIM + n0 + row];
             b.y = B1[(size_t)(ka + 1) * FDIM + n0 + row];
      acc1 = __builtin_amdgcn_wmma_f32_16x16x4_f32(false, a, false, b,
                                                   (short)0, acc1, false, false);
    }
    if (NB > 2) {
      v2f b; b.x = B2[(size_t)ka * FDIM + n0 + row];
             b.y = B2[(size_t)(ka + 1) * FDIM + n0 + row];
      acc2 = __builtin_amdgcn_wmma_f32_16x16x4_f32(false, a, false, b,
                                                   (short)0, acc2, false, false);
    }
  }

#pragma unroll
  for (int r = 0; r < 8; ++r) {
    const int m = m0 + r + half * 8;
    if (m < M) {
      const size_t off = (size_t)m * FDIM + n0 + row;
      C0[off] = acc0[r];
      if (NB > 1) C1[off] = acc1[r];
      if (NB > 2) C2[off] = acc2[r];
    }
  }
}

// ---------------------------------------------------------------------------
// q_t[128] = time_embedding[idx_t] @ WQ[128:,:]   (single 128-vector)
// ---------------------------------------------------------------------------
__global__ __launch_bounds__(128) void qt_kernel(const float* __restrict__ t,
                                                 const float* __restrict__ temb,
                                                 const float* __restrict__ WQ,
                                                 float* __restrict__ qt) {
  const int o = threadIdx.x;
  int idx = (int)floorf(t[0] / DELTAT);
  idx = min(max(idx, 0), TROWS - 1);
  const float* __restrict__ phi = temb + (size_t)idx * FDIM;
  float acc = 0.0f;
#pragma unroll 8
  for (int d = 0; d < FDIM; ++d)
    acc = fmaf(phi[d], WQ[(size_t)(FDIM + d) * FDIM + o], acc);
  qt[o] = acc;
}

// ---------------------------------------------------------------------------
// Pass A: per-edge logit alpha = dot(XQ[dst]+qt, XK[src]+TK[idx_ts]),
//          atomicMax into per-dst running max. One wave per edge (4 f32/lane).
// ---------------------------------------------------------------------------
__global__ __launch_bounds__(256) void edge_alpha_kernel(
    const int* __restrict__ ei, const float* __restrict__ ts,
    const float* __restrict__ XQ, const float* __restrict__ XK,
    const float* __restrict__ TK, const float* __restrict__ qt,
    float* __restrict__ alpha, int* __restrict__ idxbuf,
    unsigned* __restrict__ menc) {
  const int e = blockIdx.x * 8 + (threadIdx.x >> 5);
  const int lane = threadIdx.x & 31;
  if (e >= NEDGES) return;

  const int src = ei[e];
  const int dst = ei[NEDGES + e];
  int idx = (int)floorf(ts[e] / DELTAT);
  idx = min(max(idx, 0), TROWS - 1);

  const float4 q0 = reinterpret_cast<const float4*>(XQ + (size_t)dst * FDIM)[lane];
  const float4 qb = reinterpret_cast<const float4*>(qt)[lane];
  const float4 k0 = reinterpret_cast<const float4*>(XK + (size_t)src * FDIM)[lane];
  const float4 kb = reinterpret_cast<const float4*>(TK + (size_t)idx * FDIM)[lane];

  float p = (q0.x + qb.x) * (k0.x + kb.x) + (q0.y + qb.y) * (k0.y + kb.y) +
            (q0.z + qb.z) * (k0.z + kb.z) + (q0.w + qb.w) * (k0.w + kb.w);
#pragma unroll
  for (int o = 16; o > 0; o >>= 1) p += __shfl_xor(p, o, 32);

  if (lane == 0) {
    alpha[e]  = p;
    idxbuf[e] = idx;
    atomicMax(menc + dst, fenc(p));
  }
}

// ---------------------------------------------------------------------------
// Pass B: e = exp(alpha - m[dst]); atomicAdd into per-dst sum.
// ---------------------------------------------------------------------------
__global__ __launch_bounds__(256) void edge_exp_kernel(
    const int* __restrict__ ei, const float* __restrict__ alpha,
    const unsigned* __restrict__ menc, float* __restrict__ ex,
    float* __restrict__ s) {
  const int e = blockIdx.x * 256 + threadIdx.x;
  if (e >= NEDGES) return;
  const int dst = ei[NEDGES + e];
  const float m = fdec(menc[dst]);   // dst has >=1 edge, so encoding is valid
  const float v = expf(alpha[e] - m);
  ex[e] = v;
  unsafeAtomicAdd(s + dst, v);
}

// ---------------------------------------------------------------------------
// Pass C: out[dst] += (XV[src] + TV[idx]) * (e / (s[dst] + 1e-16))
// One wave per edge; 4 hardware fp32 atomics per lane into L2-resident out.
// ---------------------------------------------------------------------------
__global__ __launch_bounds__(256) void edge_scatter_kernel(
    const int* __restrict__ ei, const int* __restrict__ idxbuf,
    const float* __restrict__ ex, const float* __restrict__ s,
    const float* __restrict__ XV, const float* __restrict__ TV,
    float* __restrict__ out) {
  const int e = blockIdx.x * 8 + (threadIdx.x >> 5);
  const int lane = threadIdx.x & 31;
  if (e >= NEDGES) return;

  const int src = ei[e];
  const int dst = ei[NEDGES + e];
  const int idx = idxbuf[e];
  const float a = ex[e] / (s[dst] + 1e-16f);

  const float4 v0 = reinterpret_cast<const float4*>(XV + (size_t)src * FDIM)[lane];
  const float4 vb = reinterpret_cast<const float4*>(TV + (size_t)idx * FDIM)[lane];

  float* o = out + (size_t)dst * FDIM + lane * 4;
  unsafeAtomicAdd(o + 0, (v0.x + vb.x) * a);
  unsafeAtomicAdd(o + 1, (v0.y + vb.y) * a);
  unsafeAtomicAdd(o + 2, (v0.z + vb.z) * a);
  unsafeAtomicAdd(o + 3, (v0.w + vb.w) * a);
}

// ---------------------------------------------------------------------------
// host launcher
// ---------------------------------------------------------------------------
extern "C" void kernel_launch(void* const* d_in, const int* in_sizes, int n_in,
                              void* d_out, int out_size, void* d_ws, size_t ws_size,
                              hipStream_t stream) {
  const float* x    = (const float*)d_in[0];   // [N,128]
  const int*   ei   = (const int*)d_in[1];     // [2,E] flat (row0=src, row1=dst)
  const float* ts   = (const float*)d_in[2];   // [E]
  const float* t    = (const float*)d_in[3];   // [1]
  const float* temb = (const float*)d_in[4];   // [ROWS,128]
  const float* WQ   = (const float*)d_in[5];   // [256,128]
  const float* WK   = (const float*)d_in[6];
  const float* WV   = (const float*)d_in[7];
  float* out = (float*)d_out;                  // [N,128]

  // workspace carve-up (~95 MB total)
  float* ws    = (float*)d_ws;
  float* XQ    = ws;                                   // N*128
  float* XK    = XQ + (size_t)NNODES * FDIM;           // N*128
  float* XV    = XK + (size_t)NNODES * FDIM;           // N*128
  float* TK    = XV + (size_t)NNODES * FDIM;           // ROWS*128
  float* TV    = TK + (size_t)TROWS * FDIM;            // ROWS*128
  float* qt    = TV + (size_t)TROWS * FDIM;            // 128
  float* alpha = qt + FDIM;                            // E
  float* ex    = alpha + NEDGES;                       // E
  float* s     = ex + NEDGES;                          // N
  unsigned* menc = (unsigned*)(s + NNODES);            // N
  int* idxbuf    = (int*)(menc + NNODES);              // E

  init_kernel<<<(NNODES * FDIM + 255) / 256, 256, 0, stream>>>(out, s, menc);

  // node projections: XQ/XK/XV = x @ W*[0:128,:]   (fused, x read once)
  gemm_wmma_k128<3><<<(NNODES + 15) / 16, 256, 0, stream>>>(
      x, NNODES, WQ, WK, WV, XQ, XK, XV);

  // time-table projections: TK/TV = T @ W*[128:256,:]
  gemm_wmma_k128<2><<<TROWS / 16, 256, 0, stream>>>(
      temb, TROWS, WK + FDIM * FDIM, WV + FDIM * FDIM, nullptr, TK, TV, nullptr);

  // single query-time vector
  qt_kernel<<<1, FDIM, 0, stream>>>(t, temb, WQ, qt);

  // edge phase: logits+max, exp+sum, weighted scatter
  edge_alpha_kernel<<<(NEDGES + 7) / 8, 256, 0, stream>>>(
      ei, ts, XQ, XK, TK, qt, alpha, idxbuf, menc);
  edge_exp_kernel<<<(NEDGES + 255) / 256, 256, 0, stream>>>(
      ei, alpha, menc, ex, s);
  edge_scatter_kernel<<<(NEDGES + 7) / 8, 256, 0, stream>>>(
      ei, idxbuf, ex, s, XV, TV, out);
}